// SGBlock_26371099198196
// MI455X (gfx1250) — compile-verified
//
#include <hip/hip_runtime.h>
#include <math.h>

typedef __attribute__((ext_vector_type(2))) float v2f;
typedef __attribute__((ext_vector_type(8))) float v8f;

#define DIMX 768
#define HID  8
#define NTOK 320
#define NB   512

// ---------------------------------------------------------------------------
// Kernel A0: zero-pad adw (8x768) -> adwp (16x768, rows 8..15 = 0)
// Removes the divergent guard from kernel A's inner loop.
// ---------------------------------------------------------------------------
__global__ __launch_bounds__(256)
void sg_pad_adw(const float* __restrict__ adw, float* __restrict__ adwp)
{
    int i = blockIdx.x * 256 + threadIdx.x;   // 0 .. 16*768-1
    if (i >= 16 * DIMX) return;
    int o = i / DIMX;
    adwp[i] = (o < 8) ? adw[i] : 0.0f;
}

// ---------------------------------------------------------------------------
// Kernel A: adapter_down  h[row][o] = sum_d x[row][d]*adw[o][d] + adb[o]
// One wave per 16-row tile. D(16x16) = A(16x4: adwp) x B(4x16: x rows).
// Layouts per CDNA5 ISA 7.12.2:
//   A vgpr0 = K0 (lanes0-15) / K2 (lanes16-31), vgpr1 = K1 / K3  (M = lane%16)
//   B vgpr0 = B[K0][n] / B[K2][n], vgpr1 = B[K1][n] / B[K3][n]   (N = lane%16)
//   D vgpr g: lanes0-15 -> D[m=g][n=lane]  (only half=0, m<8 is stored)
// Software-pipelined: loads for step i+1 issue before WMMA of step i.
// ---------------------------------------------------------------------------
__global__ __launch_bounds__(128)
void sg_adapter_down_wmma(const float* __restrict__ x,
                          const float* __restrict__ adwp,
                          const float* __restrict__ adb,
                          float* __restrict__ h)
{
    const int wave = threadIdx.x >> 5;
    const int lane = threadIdx.x & 31;
    const int half = lane >> 4;
    const int lr   = lane & 15;
    const long tile = (long)blockIdx.x * 4 + wave;
    const long row0 = tile * 16;

    const float* xrow = x    + (row0 + lr) * DIMX + 2 * half;  // B operand
    const float* arow = adwp + (long)lr   * DIMX + 2 * half;   // A operand (padded)

    v8f acc = {};

    v2f a0, b0;
    a0.x = arow[0]; a0.y = arow[1];
    b0.x = xrow[0]; b0.y = xrow[1];

#pragma unroll 4
    for (int k0 = 0; k0 < DIMX - 4; k0 += 4) {
        v2f a1, b1;
        a1.x = arow[k0 + 4]; a1.y = arow[k0 + 5];
        b1.x = xrow[k0 + 4]; b1.y = xrow[k0 + 5];
        // 8 args: (neg_a, A, neg_b, B, c_mod, C, reuse_a, reuse_b)
        acc = __builtin_amdgcn_wmma_f32_16x16x4_f32(
            false, a0, false, b0, (short)0, acc, false, false);
        a0 = a1;
        b0 = b1;
    }
    acc = __builtin_amdgcn_wmma_f32_16x16x4_f32(
        false, a0, false, b0, (short)0, acc, false, false);

    if (half == 0) {
        float* hr = h + (row0 + lr) * HID;
#pragma unroll
        for (int g = 0; g < 8; ++g) hr[g] = acc[g] + adb[g];
    }
}

// ---------------------------------------------------------------------------
// Kernel B: middle (per-batch-element).  All in LDS; direct DFTs (A<=16).
// ---------------------------------------------------------------------------
template <int A, int KF, int TOFF>
__device__ void sg_branch(const float* __restrict__ s_h,
                          float* s_x1, float* s_tre, float* s_tim,
                          float* s_fre, float* s_fim, float* s_out,
                          const float* bw, const float* bb,
                          const float* bng, const float* bnb,
                          const float* rw, const float* rb,
                          const float* iw, const float* ib,
                          const float* n1g, const float* n1b,
                          const float* n2g, const float* n2b,
                          const float* cw,
                          const float* ow, const float* ob,
                          float* __restrict__ catb)
{
    const int tid = threadIdx.x;
    const int nthr = blockDim.x;
    const float rs = rsqrtf(1.0f + 1e-5f);
    const float TWO_PI = 6.283185307179586f;
    const float invA = 1.0f / (float)A;  // ortho: 1/sqrt(A*A)

    // S1: conv1x1 in + BN + ReLU -> s_x1[y*A+x][c]
    for (int e = tid; e < A * A * HID; e += nthr) {
        int c = e & 7, p = e >> 3;
        int n = TOFF + p;
        float v = bb[c];
#pragma unroll
        for (int cc = 0; cc < 8; ++cc) v += s_h[n * 8 + cc] * bw[c * 8 + cc];
        v = v * (bng[c] * rs) + bnb[c];
        s_x1[p * 8 + c] = fmaxf(v, 0.0f);
    }
    __syncthreads();

    // S2: rfft along H (y): T[x][k][c]
    for (int e = tid; e < A * KF * HID; e += nthr) {
        int c = e & 7, t = e >> 3, k = t % KF, xx = t / KF;
        float re = 0.f, im = 0.f;
        for (int y = 0; y < A; ++y) {
            float v = s_x1[(y * A + xx) * 8 + c];
            float sn, cs;
            __sincosf(-TWO_PI * (float)(y * k) * invA, &sn, &cs);
            re += v * cs;
            im += v * sn;
        }
        s_tre[(xx * KF + k) * 8 + c] = re;
        s_tim[(xx * KF + k) * 8 + c] = im;
    }
    __syncthreads();

    // S3: full fft along W (x): F[m][k][c] = (1/A) * sum_x T e^{-i2pi x m/A}
    for (int e = tid; e < A * KF * HID; e += nthr) {
        int c = e & 7, t = e >> 3, k = t % KF, m = t / KF;
        float re = 0.f, im = 0.f;
        for (int xx = 0; xx < A; ++xx) {
            float ar = s_tre[(xx * KF + k) * 8 + c];
            float ai = s_tim[(xx * KF + k) * 8 + c];
            float sn, cs;
            __sincosf(TWO_PI * (float)(xx * m) * invA, &sn, &cs);
            re += ar * cs + ai * sn;   // *(cos - i sin)
            im += ai * cs - ar * sn;
        }
        s_fre[(m * KF + k) * 8 + c] = re * invA;
        s_fim[(m * KF + k) * 8 + c] = im * invA;
    }
    __syncthreads();

    // S4: 8x8 linears on Re/Im + BN + ReLU, then complex gate -> P in s_tre/s_tim
    for (int e = tid; e < A * KF * HID; e += nthr) {
        int o = e & 7, t = e >> 3;
        float vr = rb[o], vi = ib[o];
#pragma unroll
        for (int c = 0; c < 8; ++c) {
            vr += s_fre[t * 8 + c] * rw[o * 8 + c];
            vi += s_fim[t * 8 + c] * iw[o * 8 + c];
        }
        vr = fmaxf(vr * (n1g[o] * rs) + n1b[o], 0.0f);
        vi = fmaxf(vi * (n2g[o] * rs) + n2b[o], 0.0f);
        float wr = cw[(t * 8 + o) * 2 + 0];
        float wi = cw[(t * 8 + o) * 2 + 1];
        s_tre[t * 8 + o] = vr * wr - vi * wi;
        s_tim[t * 8 + o] = vr * wi + vi * wr;
    }
    __syncthreads();

    // S5: inverse fft along W (m -> x): Y[x][k][o] = sum_m P e^{+i2pi x m/A}
    for (int e = tid; e < A * KF * HID; e += nthr) {
        int o = e & 7, t = e >> 3, k = t % KF, xx = t / KF;
        float re = 0.f, im = 0.f;
        for (int m = 0; m < A; ++m) {
            float pr = s_tre[(m * KF + k) * 8 + o];
            float pi = s_tim[(m * KF + k) * 8 + o];
            float sn, cs;
            __sincosf(TWO_PI * (float)(xx * m) * invA, &sn, &cs);
            re += pr * cs - pi * sn;   // *(cos + i sin)
            im += pr * sn + pi * cs;
        }
        s_fre[(xx * KF + k) * 8 + o] = re;
        s_fim[(xx * KF + k) * 8 + o] = im;
    }
    __syncthreads();

    // S6: irfft along H (k -> y), Hermitian half-spectrum, + residual
    for (int e = tid; e < A * A * HID; e += nthr) {
        int c = e & 7, p = e >> 3, xx = p % A, y = p / A;
        float acc = s_fre[(xx * KF + 0) * 8 + c];                       // DC (Im ignored)
        acc += ((y & 1) ? -1.0f : 1.0f) * s_fre[(xx * KF + (KF - 1)) * 8 + c]; // Nyquist
        for (int k = 1; k < KF - 1; ++k) {
            float sn, cs;
            __sincosf(TWO_PI * (float)(y * k) * invA, &sn, &cs);
            acc += 2.0f * (s_fre[(xx * KF + k) * 8 + c] * cs -
                           s_fim[(xx * KF + k) * 8 + c] * sn);
        }
        s_out[p * 8 + c] = acc * invA + s_x1[p * 8 + c];
    }
    __syncthreads();

    // S7: out conv1x1 + patch2token write into cat
    for (int e = tid; e < A * A * HID; e += nthr) {
        int c = e & 7, p = e >> 3;
        float v = ob[c];
#pragma unroll
        for (int cc = 0; cc < 8; ++cc) v += s_out[p * 8 + cc] * ow[c * 8 + cc];
        catb[(TOFF + p) * 8 + c] = v;
    }
    __syncthreads();
}

__global__ __launch_bounds__(256)
void sg_middle_kernel(const float* __restrict__ h,
                      const float* b1w, const float* b1b,
                      const float* bn1g, const float* bn1b,
                      const float* b2w, const float* b2b,
                      const float* bn2g, const float* bn2b,
                      const float* sxrw, const float* sxrb,
                      const float* sxiw, const float* sxib,
                      const float* sxn1g, const float* sxn1b,
                      const float* sxn2g, const float* sxn2b,
                      const float* cwx,
                      const float* szrw, const float* szrb,
                      const float* sziw, const float* szib,
                      const float* szn1g, const float* szn1b,
                      const float* szn2g, const float* szn2b,
                      const float* cwz,
                      const float* cow, const float* cob,
                      const float* czw, const float* czb,
                      float* __restrict__ cat)
{
    __shared__ float s_h[NTOK * HID];       // 2560
    __shared__ float s_x1[16 * 16 * 8];     // 2048
    __shared__ float s_tre[16 * 9 * 8];     // 1152
    __shared__ float s_tim[16 * 9 * 8];
    __shared__ float s_fre[16 * 9 * 8];
    __shared__ float s_fim[16 * 9 * 8];
    __shared__ float s_out[16 * 16 * 8];    // 2048  -> ~45KB total

    const int b = blockIdx.x;
    const float* hb = h + (long)b * NTOK * HID;
    float* catb = cat + (long)b * NTOK * HID;

    for (int i = threadIdx.x; i < NTOK * HID; i += blockDim.x) s_h[i] = hb[i];
    __syncthreads();

    // x branch: tokens 64..319, 16x16 spatial, 9 freq bins
    sg_branch<16, 9, 64>(s_h, s_x1, s_tre, s_tim, s_fre, s_fim, s_out,
                         b1w, b1b, bn1g, bn1b,
                         sxrw, sxrb, sxiw, sxib,
                         sxn1g, sxn1b, sxn2g, sxn2b, cwx,
                         cow, cob, catb);
    // z branch: tokens 0..63, 8x8 spatial, 5 freq bins
    sg_branch<8, 5, 0>(s_h, s_x1, s_tre, s_tim, s_fre, s_fim, s_out,
                       b2w, b2b, bn2g, bn2b,
                       szrw, szrb, sziw, szib,
                       szn1g, szn1b, szn2g, szn2b, cwz,
                       czw, czb, catb);
}

// ---------------------------------------------------------------------------
// Kernel C: adapter_up  out[row][d] = sum_o cat[row][o]*auw[d][o] + aub[d]
// One float4 of output per thread; store-bandwidth bound, fully coalesced.
// ---------------------------------------------------------------------------
__global__ __launch_bounds__(256)
void sg_adapter_up(const float* __restrict__ cat,
                   const float* __restrict__ auw,
                   const float* __restrict__ aub,
                   float* __restrict__ out)
{
    const long total = (long)NB * NTOK * (DIMX / 4);
    long idx = (long)blockIdx.x * blockDim.x + threadIdx.x;
    if (idx >= total) return;
    long row = idx / (DIMX / 4);
    int dq = (int)(idx % (DIMX / 4));
    int d = dq * 4;

    const float* cr = cat + row * HID;
    float c0 = cr[0], c1 = cr[1], c2 = cr[2], c3 = cr[3];
    float c4 = cr[4], c5 = cr[5], c6 = cr[6], c7 = cr[7];

    float4 r;
    float* rp = &r.x;
#pragma unroll
    for (int j = 0; j < 4; ++j) {
        const float* wd = auw + (long)(d + j) * HID;
        float v = aub[d + j];
        v += c0 * wd[0] + c1 * wd[1] + c2 * wd[2] + c3 * wd[3];
        v += c4 * wd[4] + c5 * wd[5] + c6 * wd[6] + c7 * wd[7];
        rp[j] = v;
    }
    reinterpret_cast<float4*>(out)[idx] = r;
}

// ---------------------------------------------------------------------------
extern "C" void kernel_launch(void* const* d_in, const int* in_sizes, int n_in,
                              void* d_out, int out_size, void* d_ws, size_t ws_size,
                              hipStream_t stream)
{
    const float* x     = (const float*)d_in[0];
    const float* adw   = (const float*)d_in[1];
    const float* adb   = (const float*)d_in[2];
    const float* auw   = (const float*)d_in[3];
    const float* aub   = (const float*)d_in[4];
    const float* b1w   = (const float*)d_in[5];
    const float* b1b   = (const float*)d_in[6];
    const float* bn1g  = (const float*)d_in[7];
    const float* bn1b  = (const float*)d_in[8];
    const float* b2w   = (const float*)d_in[9];
    const float* b2b   = (const float*)d_in[10];
    const float* bn2g  = (const float*)d_in[11];
    const float* bn2b  = (const float*)d_in[12];
    const float* sxrw  = (const float*)d_in[13];
    const float* sxrb  = (const float*)d_in[14];
    const float* sxiw  = (const float*)d_in[15];
    const float* sxib  = (const float*)d_in[16];
    const float* sxn1g = (const float*)d_in[17];
    const float* sxn1b = (const float*)d_in[18];
    const float* sxn2g = (const float*)d_in[19];
    const float* sxn2b = (const float*)d_in[20];
    const float* cwx   = (const float*)d_in[21];
    const float* szrw  = (const float*)d_in[22];
    const float* szrb  = (const float*)d_in[23];
    const float* sziw  = (const float*)d_in[24];
    const float* szib  = (const float*)d_in[25];
    const float* szn1g = (const float*)d_in[26];
    const float* szn1b = (const float*)d_in[27];
    const float* szn2g = (const float*)d_in[28];
    const float* szn2b = (const float*)d_in[29];
    const float* cwz   = (const float*)d_in[30];
    const float* cow   = (const float*)d_in[31];
    const float* cob   = (const float*)d_in[32];
    const float* czw   = (const float*)d_in[33];
    const float* czb   = (const float*)d_in[34];

    float* h    = (float*)d_ws;                        // 512*320*8 floats
    float* cat  = h + (size_t)NB * NTOK * HID;         // 512*320*8 floats
    float* adwp = cat + (size_t)NB * NTOK * HID;       // 16*768 floats

    // A0: zero-pad adw into 16x768
    sg_pad_adw<<<(16 * DIMX + 255) / 256, 256, 0, stream>>>(adw, adwp);

    // A: 163840 rows / 16 per wave / 4 waves per block = 2560 blocks
    sg_adapter_down_wmma<<<2560, 128, 0, stream>>>(x, adwp, adb, h);

    // B: one workgroup per batch element
    sg_middle_kernel<<<NB, 256, 0, stream>>>(h,
        b1w, b1b, bn1g, bn1b, b2w, b2b, bn2g, bn2b,
        sxrw, sxrb, sxiw, sxib, sxn1g, sxn1b, sxn2g, sxn2b, cwx,
        szrw, szrb, sziw, szib, szn1g, szn1b, szn2g, szn2b, cwz,
        cow, cob, czw, czb, cat);

    // C: float4 outputs
    const long totalC = (long)NB * NTOK * (DIMX / 4);
    const int blocksC = (int)((totalC + 255) / 256);
    sg_adapter_up<<<blocksC, 256, 0, stream>>>(cat, auw, aub, (float*)d_out);
}